// BaselineModel_53455162966068
// MI455X (gfx1250) — compile-verified
//
#include <hip/hip_runtime.h>
#include <hip/hip_bf16.h>
#include <math.h>

// ---------------- problem constants ----------------
#define Bsz   32
#define Lseq  512
#define Hdim  1024
#define Tnum  67
#define TP    80      // N padded to 5 tiles of 16
#define EP    68      // emission row stride (floats)
#define HP    68      // history row stride (bytes), 4-aligned
#define KC    64      // K chunk per LDS stage
#define ROWS  128     // M rows per block in GEMM kernel
#define SA    68      // LDS A stride (floats)  -> conflict-free b64 frag reads
#define SBT   68      // LDS B^T stride (floats) -> conflict-free b64 frag reads
#define SZ    81      // LDS z stride (floats)  -> conflict-free row scans

typedef float v2f __attribute__((ext_vector_type(2)));
typedef float v8f __attribute__((ext_vector_type(8)));

// =====================================================================
// Kernel 1: emissions = log_softmax(leaky_relu(hidden) @ W + b)
// fp32 WMMA (V_WMMA_F32_16X16X4_F32). One wave owns one 16-row M tile
// and all five 16-col N tiles (5 x v8f accumulators). A and B^T staged
// in LDS so every fragment is a single conflict-free ds_load_b64.
// =====================================================================
__global__ __launch_bounds__(256)
void k1_emissions(const float* __restrict__ hidden,
                  const float* __restrict__ W,
                  const float* __restrict__ bias,
                  float* __restrict__ emis)
{
    __shared__ float smem[ROWS * SA + TP * SBT];   // 14144 floats = 56.6 KB
    float* As  = smem;                             // [128][SA]
    float* BsT = smem + ROWS * SA;                 // [80][SBT]  (W^T tile)
    float* zs  = smem;                             // [128][SZ] aliased later

    const int tid  = threadIdx.x;
    const int lane = tid & 31;
    const int wv   = tid >> 5;           // wave id 0..7 -> M tile
    const int lo   = lane & 15;          // row-in-tile (A) / col-in-tile (B,C)
    const int hi   = lane >> 4;          // half-wave K/M group select
    const int row0 = blockIdx.x * ROWS;

    v8f acc[5];
#pragma unroll
    for (int nt = 0; nt < 5; ++nt) acc[nt] = 0.0f;

    // per-wave fragment base pointers (fixed; kk becomes an immediate offset)
    const float* Arow = As + (16 * wv + lo) * SA + 2 * hi;
    const float* Brow = BsT + lo * SBT + 2 * hi;

    for (int kc = 0; kc < Hdim; kc += KC) {
        __syncthreads();
        // ---- stage A (128 x 64) with leaky_relu, coalesced float4 ----
        {
            int r  = tid >> 4;           // 0..15
            int c4 = (tid & 15) * 4;     // 0..60
#pragma unroll
            for (int p = 0; p < 8; ++p, r += 16) {
                const float* src = hidden + (size_t)(row0 + r) * Hdim + kc + c4;
                float4 v = *reinterpret_cast<const float4*>(src);
                v.x = (v.x > 0.f) ? v.x : 0.01f * v.x;
                v.y = (v.y > 0.f) ? v.y : 0.01f * v.y;
                v.z = (v.z > 0.f) ? v.z : 0.01f * v.z;
                v.w = (v.w > 0.f) ? v.w : 0.01f * v.w;
                float* dst = As + r * SA + c4;
                dst[0] = v.x; dst[1] = v.y; dst[2] = v.z; dst[3] = v.w;
            }
        }
        // ---- stage B transposed: BsT[n][k] = W[kc+k][n], zero-pad n>=67 ----
        for (int idx = tid; idx < KC * TP; idx += 256) {
            int k = idx / TP, n = idx % TP;               // coalesced W reads
            BsT[n * SBT + k] = (n < Tnum) ? W[(size_t)(kc + k) * Tnum + n] : 0.0f;
        }
        __syncthreads();
        // ---- WMMA inner loop: 16 K-steps of 4 ----
#pragma unroll 4
        for (int kk = 0; kk < KC; kk += 4) {
            // A frag: reg0 = A[m][kk+2hi], reg1 = A[m][kk+2hi+1]
            v2f a = *reinterpret_cast<const v2f*>(Arow + kk);
            // B frags: reg0 = B[kk+2hi][n], reg1 = B[kk+2hi+1][n]
            v2f bf[5];
#pragma unroll
            for (int nt = 0; nt < 5; ++nt)
                bf[nt] = *reinterpret_cast<const v2f*>(Brow + nt * 16 * SBT + kk);
#pragma unroll
            for (int nt = 0; nt < 5; ++nt)
                acc[nt] = __builtin_amdgcn_wmma_f32_16x16x4_f32(
                    false, a, false, bf[nt], (short)0, acc[nt], false, false);
        }
    }
    __syncthreads();
    // ---- epilogue: bias add, spill z to LDS (row m = 16w + v + 8*hi) ----
#pragma unroll
    for (int nt = 0; nt < 5; ++nt) {
        int n = nt * 16 + lo;
        float bn = (n < Tnum) ? bias[n] : 0.0f;
#pragma unroll
        for (int v = 0; v < 8; ++v) {
            int r = 16 * wv + v + 8 * hi;
            zs[r * SZ + n] = acc[nt][v] + bn;
        }
    }
    __syncthreads();
    // ---- log_softmax over the 67 real classes, write emissions ----
    if (tid < ROWS) {
        const float* zr = zs + tid * SZ;
        float m = -INFINITY;
        for (int j = 0; j < Tnum; ++j) m = fmaxf(m, zr[j]);
        float s = 0.f;
        for (int j = 0; j < Tnum; ++j) s += __expf(zr[j] - m);
        float lse = m + __logf(s);
        float* out = emis + (size_t)(row0 + tid) * EP;
        for (int j = 0; j < Tnum; ++j) out[j] = zr[j] - lse;
    }
}

// =====================================================================
// Kernel 2: CRF forward (logsumexp scan) + Viterbi forward, one block
// per batch element. transT in LDS (stride 69 -> conflict-free),
// alpha/score double-buffered, one barrier per time step.
// =====================================================================
__global__ __launch_bounds__(128)
void k2_crf_forward(const float* __restrict__ emis,
                    const float* __restrict__ startT,
                    const float* __restrict__ endT,
                    const float* __restrict__ trans,
                    const int* __restrict__ labels,
                    const unsigned char* __restrict__ mask,
                    float* __restrict__ nd,
                    int* __restrict__ lastIdx,
                    unsigned char* __restrict__ hist)
{
    __shared__ float transT[Tnum * 69];
    __shared__ float alpha[2][Tnum];
    __shared__ float scoreV[2][Tnum];
    __shared__ float red[3 * 128];
    __shared__ float resNum;

    const int b   = blockIdx.x;
    const int tid = threadIdx.x;
    const float* eb = emis + (size_t)b * Lseq * EP;
    const unsigned char* mb = mask + (size_t)b * Lseq;
    const int* lb = labels + (size_t)b * Lseq;

    for (int idx = tid; idx < Tnum * Tnum; idx += 128) {
        int i = idx / Tnum, j = idx % Tnum;
        transT[j * 69 + i] = trans[idx];          // transposed
    }
    if (tid < Tnum) {
        float e0 = eb[tid];
        alpha[0][tid]  = startT[tid] + e0;
        scoreV[0][tid] = startT[tid] + e0;
    }
    // ---- numerator partial sums (parallel over t) ----
    float emit_s = 0.f, tr_s = 0.f, len_s = 0.f;
    for (int t = tid; t < Lseq; t += 128) {
        if (mb[t]) {
            len_s  += 1.f;
            emit_s += eb[t * EP + lb[t]];
            if (t >= 1) tr_s += trans[lb[t - 1] * Tnum + lb[t]];
        }
    }
    red[tid] = emit_s; red[128 + tid] = tr_s; red[256 + tid] = len_s;
    __syncthreads();
    if (tid == 0) {
        float es = 0.f, ts = 0.f, ls = 0.f;
        for (int i = 0; i < 128; ++i) { es += red[i]; ts += red[128 + i]; ls += red[256 + i]; }
        int len = (int)(ls + 0.5f);
        resNum = startT[lb[0]] + es + ts + endT[lb[len - 1]];
    }
    __syncthreads();

    // ---- sequential scan over time ----
    int cur = 0;
    for (int t = 1; t < Lseq; ++t) {
        const int nxt = cur ^ 1;
        const int mt  = (int)mb[t];
        if (tid < Tnum) {
            const int j = tid;
            const float* tr_j = transT + j * 69;
            float m = -INFINITY;
            float bestv = -INFINITY; int besti = 0;
            for (int i = 0; i < Tnum; ++i) {
                float tv = tr_j[i];
                float a  = alpha[cur][i] + tv;
                m = fmaxf(m, a);
                float sc = scoreV[cur][i] + tv;
                if (sc > bestv) { bestv = sc; besti = i; }   // first-max like argmax
            }
            float s = 0.f;
            for (int i = 0; i < Tnum; ++i)
                s += __expf(alpha[cur][i] + tr_j[i] - m);
            float e_tj = eb[t * EP + j];
            float na   = m + __logf(s) + e_tj;
            float nsc  = bestv + e_tj;
            alpha[nxt][j]  = mt ? na  : alpha[cur][j];
            scoreV[nxt][j] = mt ? nsc : scoreV[cur][j];
            hist[((size_t)(t - 1) * Bsz + b) * HP + j] = (unsigned char)besti;
        }
        __syncthreads();
        cur = nxt;
    }

    // ---- denominator (logsumexp(alpha+end)) and viterbi last state ----
    if (tid == 0) {
        float m = -INFINITY;
        for (int j = 0; j < Tnum; ++j) m = fmaxf(m, alpha[cur][j] + endT[j]);
        float s = 0.f;
        for (int j = 0; j < Tnum; ++j) s += __expf(alpha[cur][j] + endT[j] - m);
        nd[b] = resNum - (m + __logf(s));
        float bv = -INFINITY; int bi = 0;
        for (int j = 0; j < Tnum; ++j) {
            float v = scoreV[cur][j] + endT[j];
            if (v > bv) { bv = v; bi = j; }
        }
        lastIdx[b] = bi;
    }
}

// =====================================================================
// Kernel 3: Viterbi backtrack through LDS-staged history + loss scalar
// =====================================================================
__global__ __launch_bounds__(128)
void k3_backtrack(const unsigned char* __restrict__ hist,
                  const int* __restrict__ lastIdx,
                  const unsigned char* __restrict__ mask,
                  const float* __restrict__ nd,
                  float* __restrict__ dout)
{
    __shared__ unsigned int histS[(Lseq - 1) * (HP / 4)];   // 511*17 u32
    __shared__ int tagsS[Lseq];
    const int b   = blockIdx.x;
    const int tid = threadIdx.x;
    const int rowU = HP / 4;

    const unsigned int* hg = reinterpret_cast<const unsigned int*>(hist);
    for (int idx = tid; idx < (Lseq - 1) * rowU; idx += 128) {
        int k = idx / rowU, u = idx % rowU;
        histS[idx] = hg[((size_t)k * Bsz + b) * rowU + u];
    }
    __syncthreads();

    const unsigned char* hb = reinterpret_cast<const unsigned char*>(histS);
    const unsigned char* mb = mask + (size_t)b * Lseq;
    if (tid == 0) {
        int cur = lastIdx[b];
        for (int k = Lseq - 2; k >= 0; --k) {
            tagsS[k + 1] = cur;                  // record before update
            if (mb[k + 1]) cur = (int)hb[k * HP + cur];
        }
        tagsS[0] = cur;
    }
    __syncthreads();

    float* outTags = dout + 1;
    for (int t = tid; t < Lseq; t += 128) {
        int tag = mb[t] ? tagsS[t] : 0;          // where(mask, tags, 0)
        outTags[(size_t)b * Lseq + t] = (float)tag;
    }
    if (b == 0 && tid == 0) {
        float s = 0.f;
        for (int i = 0; i < Bsz; ++i) s += nd[i];
        dout[0] = -s / (float)Bsz;               // loss = -llh / B
    }
}

// =====================================================================
extern "C" void kernel_launch(void* const* d_in, const int* in_sizes, int n_in,
                              void* d_out, int out_size, void* d_ws, size_t ws_size,
                              hipStream_t stream)
{
    const float* hidden = (const float*)d_in[0];
    const float* W      = (const float*)d_in[1];
    const float* bias   = (const float*)d_in[2];
    const float* startT = (const float*)d_in[3];
    const float* endT   = (const float*)d_in[4];
    const float* trans  = (const float*)d_in[5];
    const int*   labels = (const int*)d_in[6];
    const unsigned char* mask = (const unsigned char*)d_in[7];

    char* ws = (char*)d_ws;
    float* emis = (float*)ws;                                     // 16384*68 f32
    size_t off = (size_t)Bsz * Lseq * EP * sizeof(float);
    float* nd = (float*)(ws + off);        off += Bsz * sizeof(float);
    int* lastIdx = (int*)(ws + off);       off += Bsz * sizeof(int);
    unsigned char* hist = (unsigned char*)(ws + off);             // 511*32*68 B

    hipLaunchKernelGGL(k1_emissions, dim3((Bsz * Lseq) / ROWS), dim3(256), 0, stream,
                       hidden, W, bias, emis);
    hipLaunchKernelGGL(k2_crf_forward, dim3(Bsz), dim3(128), 0, stream,
                       emis, startT, endT, trans, labels, mask, nd, lastIdx, hist);
    hipLaunchKernelGGL(k3_backtrack, dim3(Bsz), dim3(128), 0, stream,
                       hist, lastIdx, mask, nd, (float*)d_out);
}